// LinearRegressionModel_59837484368248
// MI455X (gfx1250) — compile-verified
//
#include <hip/hip_runtime.h>
#include <stdint.h>

// Problem constants (from reference): B=256, T=16, V=1024, N = T*V = 16384.
#define VOCAB     1024
#define T_CTX     16
#define BATCH     256
#define NFEAT     (T_CTX * VOCAB)      // 16384
#define ROWS      16                   // output features per block
#define TILE_FLT  (ROWS * VOCAB)       // 16384 floats = 64 KB per buffer
#define THREADS   256

// ---------------------------------------------------------------------------
// gfx1250 async copy: global -> LDS, 16 bytes per lane, tracked by ASYNCcnt.
// VDST operand carries the LDS byte address (low 32 bits of the generic
// shared-space pointer are the LDS offset per the flat-aperture rules).
// ---------------------------------------------------------------------------
__device__ __forceinline__ void async_load_b128(uint32_t lds_byte_off,
                                                const float* gaddr) {
  asm volatile("global_load_async_to_lds_b128 %0, %1, off"
               :: "v"(lds_byte_off), "v"(gaddr)
               : "memory");
}

__global__ void __launch_bounds__(THREADS, 1)
onehot_linear_relu_kernel(const int* __restrict__ x,     // [B, T] indices
                          const float* __restrict__ W,   // [N, N] row-major
                          const float* __restrict__ bias,// [N]
                          float* __restrict__ out) {     // [B, N]
  extern __shared__ float smem[];                // 2 * TILE_FLT floats
  const int tid = threadIdx.x;                   // batch index
  const int j0  = blockIdx.x * ROWS;             // output-feature tile base

  float* buf0 = smem;
  float* buf1 = smem + TILE_FLT;

  // Preload this batch row's 16 indices (64 contiguous bytes -> 4x b128).
  int vidx[T_CTX];
  {
    const int4* xr = (const int4*)(x + tid * T_CTX);
    int4 a = xr[0], b = xr[1], c = xr[2], d = xr[3];
    vidx[0]  = a.x; vidx[1]  = a.y; vidx[2]  = a.z; vidx[3]  = a.w;
    vidx[4]  = b.x; vidx[5]  = b.y; vidx[6]  = b.z; vidx[7]  = b.w;
    vidx[8]  = c.x; vidx[9]  = c.y; vidx[10] = c.z; vidx[11] = c.w;
    vidx[12] = d.x; vidx[13] = d.y; vidx[14] = d.z; vidx[15] = d.w;
  }

  float acc[ROWS];
#pragma unroll
  for (int j = 0; j < ROWS; ++j) acc[j] = 0.0f;

  // Stage tile W[j0 : j0+ROWS, t*VOCAB : (t+1)*VOCAB] into LDS buffer `dst`.
  // 4096 float4 slots, 256 threads -> 16 async b128 issues per thread
  // (k iteration == row k of the tile; thread covers float4 slot `tid`).
  auto issue_tile = [&](int t, float* dst) {
    const float*   gbase = W + (size_t)j0 * NFEAT + (size_t)t * VOCAB;
    const uint32_t lbase = (uint32_t)(uintptr_t)dst + (uint32_t)tid * 16u;
#pragma unroll
    for (int k = 0; k < ROWS; ++k) {
      const float* g = gbase + (size_t)k * NFEAT + tid * 4;
      async_load_b128(lbase + (uint32_t)k * (VOCAB * 4u), g);
    }
  };

  issue_tile(0, buf0);

#pragma unroll 1
  for (int t = 0; t < T_CTX; ++t) {
    // Prefetch next tile into the other buffer, then wait until only the
    // just-issued batch (16 per wave) is outstanding -> current tile landed
    // (async loads complete in order within a wave).
    if (t + 1 < T_CTX) {
      issue_tile(t + 1, (t & 1) ? buf0 : buf1);
      asm volatile("s_wait_asynccnt 0x10" ::: "memory");   // <= 16 in flight
    } else {
      asm volatile("s_wait_asynccnt 0x0" ::: "memory");    // all done
    }
    __syncthreads();   // everyone's async writes for tile t are in LDS

    const float* buf = (t & 1) ? buf1 : buf0;
    const int v = vidx[t];                       // x[b, t] in [0, VOCAB)
#pragma unroll
    for (int j = 0; j < ROWS; ++j)
      acc[j] += buf[j * VOCAB + v];

    __syncthreads();   // protect this buffer before next prefetch overwrites
  }

  // Epilogue: bias + ReLU. Each thread writes 16 contiguous floats at
  // out[b, j0 .. j0+15] as 4x float4 (64B contiguous per thread).
  float*       o  = out  + (size_t)tid * NFEAT + j0;
  const float* bz = bias + j0;
#pragma unroll
  for (int j = 0; j < ROWS; j += 4) {
    float4 r;
    r.x = fmaxf(acc[j + 0] + bz[j + 0], 0.0f);
    r.y = fmaxf(acc[j + 1] + bz[j + 1], 0.0f);
    r.z = fmaxf(acc[j + 2] + bz[j + 2], 0.0f);
    r.w = fmaxf(acc[j + 3] + bz[j + 3], 0.0f);
    *(float4*)(o + j) = r;
  }
}

extern "C" void kernel_launch(void* const* d_in, const int* in_sizes, int n_in,
                              void* d_out, int out_size, void* d_ws, size_t ws_size,
                              hipStream_t stream) {
  (void)in_sizes; (void)n_in; (void)d_ws; (void)ws_size; (void)out_size;
  const int*   x    = (const int*)d_in[0];    // [256, 16]
  const float* W    = (const float*)d_in[1];  // [16384, 16384]
  const float* bias = (const float*)d_in[2];  // [16384]
  float*       out  = (float*)d_out;          // [256, 16, 1024] flat

  const size_t shmem = 2u * TILE_FLT * sizeof(float);   // 128 KB -> 2 blocks/WGP
  (void)hipFuncSetAttribute((const void*)onehot_linear_relu_kernel,
                            hipFuncAttributeMaxDynamicSharedMemorySize,
                            (int)shmem);

  dim3 grid(NFEAT / ROWS);   // 1024 blocks
  dim3 block(THREADS);       // 256 threads = 8 waves (wave32)
  hipLaunchKernelGGL(onehot_linear_relu_kernel, grid, block, shmem, stream,
                     x, W, bias, out);
}